// TimeSeqMultiHeadAttention_27075473834616
// MI455X (gfx1250) — compile-verified
//
#include <hip/hip_runtime.h>
#include <hip/hip_bf16.h>

typedef float v2f __attribute__((ext_vector_type(2)));
typedef float v8f __attribute__((ext_vector_type(8)));
typedef int   v4i __attribute__((ext_vector_type(4)));

#define EMB   1024
#define SEQ   5
#define NBAT  8192
#define ROWS  (NBAT * SEQ)   // 40960
#define BM    128
#define BN    128
#define KBLK  32
#define LDP   36             // padded LDS row stride (floats): gcd(36,64)=4 -> conflict-free b64 reads
#define EPS   1e-5f

typedef __attribute__((address_space(1))) v4i GV4;   // global int4
typedef __attribute__((address_space(3))) v4i LV4;   // LDS int4

// 16-byte global -> LDS async copy (CDNA5 GLOBAL_LOAD_ASYNC_TO_LDS_B128, ASYNCcnt)
__device__ __forceinline__ void async_cp16(const float* g, float* l) {
#if __has_builtin(__builtin_amdgcn_global_load_async_to_lds_b128)
    GV4* gs = (GV4*)((v4i*)(void*)(__SIZE_TYPE__)g);   // drop const, generic -> AS1
    LV4* ld = (LV4*)((v4i*)l);                          // generic -> AS3
    __builtin_amdgcn_global_load_async_to_lds_b128(gs, ld, 0, 0);
#else
    unsigned lds_off = (unsigned)(__SIZE_TYPE__)((LV4*)(v4i*)l);
    asm volatile("global_load_async_to_lds_b128 %0, %1, off"
                 :: "v"(lds_off), "v"(g)
                 : "memory");
#endif
}

__device__ __forceinline__ void wait_async0() {
#if __has_builtin(__builtin_amdgcn_s_wait_asynccnt)
    __builtin_amdgcn_s_wait_asynccnt(0);
#else
    asm volatile("s_wait_asynccnt 0x0" ::: "memory");
#endif
}

// ---------------------------------------------------------------------------
// Kernel 1: fused QKV projection GEMM, f32 WMMA (16x16x4), async double-buffered.
// y[r,e] = sum_d x[r,d] * W[e,d] + b[e]   (torch Linear semantics)
// grid.x: M blocks (40960/128 = 320), grid.y: 24 = 3 matrices * (1024/128)
// ---------------------------------------------------------------------------
__global__ __launch_bounds__(256) void qkv_gemm_kernel(
    const float* __restrict__ X,
    const float* __restrict__ Wq, const float* __restrict__ Wk, const float* __restrict__ Wv,
    const float* __restrict__ bq, const float* __restrict__ bk, const float* __restrict__ bv,
    float* __restrict__ qo, float* __restrict__ ko, float* __restrict__ vo)
{
    __shared__ float Xl[2][BM * LDP];
    __shared__ float Wl[2][BN * LDP];

    const int tid  = threadIdx.x;
    const int lane = tid & 31;
    const int wave = tid >> 5;
    const int wm   = wave >> 1;   // 0..3 : 32-row slab
    const int wn   = wave & 1;    // 0..1 : 64-col slab
    const int hi   = lane >> 4;   // K-half select (A/B frag layout)
    const int l16  = lane & 15;

    const int gm0 = blockIdx.x * BM;
    const int mat = blockIdx.y >> 3;          // 0=q 1=k 2=v
    const int n0  = (blockIdx.y & 7) * BN;    // column block inside the matrix

    const float* W   = (mat == 0) ? Wq : ((mat == 1) ? Wk : Wv);
    const float* bia = (mat == 0) ? bq : ((mat == 1) ? bk : bv);
    float*       out = (mat == 0) ? qo : ((mat == 1) ? ko : vo);

    v8f acc[2][4];
#pragma unroll
    for (int mt = 0; mt < 2; ++mt)
#pragma unroll
        for (int nt = 0; nt < 4; ++nt)
#pragma unroll
            for (int e = 0; e < 8; ++e) acc[mt][nt][e] = 0.0f;

    // Async-stage X[BM x KBLK] and W[BN x KBLK] tiles for K-block `kb` into buffer `buf`
    auto stage = [&](int buf, int kb) {
#pragma unroll
        for (int it = 0; it < 4; ++it) {
            int idx = tid + it * 256;          // 0..1023
            int row = idx >> 3;                // 0..127
            int c4  = (idx & 7) << 2;          // 0,4,...,28  (16B aligned both sides)
            async_cp16(X + (size_t)(gm0 + row) * EMB + kb + c4, &Xl[buf][row * LDP + c4]);
            async_cp16(W + (size_t)(n0 + row) * EMB + kb + c4, &Wl[buf][row * LDP + c4]);
        }
    };

    stage(0, 0);
    int buf = 0;

    for (int kb = 0; kb < EMB; kb += KBLK) {
        wait_async0();        // my copies into `buf` have landed in LDS
        __syncthreads();      // everyone's copies landed; prior readers of buf^1 are done

        if (kb + KBLK < EMB) stage(buf ^ 1, kb + KBLK);   // overlap next tile with compute

#pragma unroll
        for (int kk = 0; kk < KBLK; kk += 4) {
            // A frag: lane(m=l16) holds X[m][kk+2*hi], X[m][kk+2*hi+1]  -> one b64 LDS load
            v2f af[2];
#pragma unroll
            for (int mt = 0; mt < 2; ++mt)
                af[mt] = *(const v2f*)(&Xl[buf][(wm * 32 + mt * 16 + l16) * LDP + kk + 2 * hi]);
            // B frag: lane(n=l16) holds W[n][kk+2*hi], W[n][kk+2*hi+1]  (B[k][n] = W[n][k])
            v2f bf[4];
#pragma unroll
            for (int nt = 0; nt < 4; ++nt)
                bf[nt] = *(const v2f*)(&Wl[buf][(wn * 64 + nt * 16 + l16) * LDP + kk + 2 * hi]);

#pragma unroll
            for (int mt = 0; mt < 2; ++mt)
#pragma unroll
                for (int nt = 0; nt < 4; ++nt)
                    acc[mt][nt] = __builtin_amdgcn_wmma_f32_16x16x4_f32(
                        false, af[mt], false, bf[nt], (short)0, acc[mt][nt], false, false);
        }
        buf ^= 1;
    }

    // Epilogue: add bias, store. C/D layout: VGPR e -> row (e + 8*hi), col l16.
#pragma unroll
    for (int mt = 0; mt < 2; ++mt) {
#pragma unroll
        for (int nt = 0; nt < 4; ++nt) {
            int nc = n0 + wn * 64 + nt * 16 + l16;
            float bb = bia[nc];
#pragma unroll
            for (int e = 0; e < 8; ++e) {
                int row = gm0 + wm * 32 + mt * 16 + 8 * hi + e;
                out[(size_t)row * EMB + nc] = acc[mt][nt][e] + bb;
            }
        }
    }
}

// ---------------------------------------------------------------------------
// Kernel 2: per-batch attention epilogue. One wave32 per batch.
// energy -> w_t mix -> LayerNorm(5) -> softmax(/32) -> att @ v
// ---------------------------------------------------------------------------
__global__ __launch_bounds__(256) void attn_kernel(
    const float* __restrict__ q, const float* __restrict__ k, const float* __restrict__ v,
    const float* __restrict__ w_t, const float* __restrict__ gamma, const float* __restrict__ beta,
    float* __restrict__ out)
{
    const int lane  = threadIdx.x & 31;
    const int batch = blockIdx.x * 8 + (threadIdx.x >> 5);
    if (batch >= NBAT) return;

    const float* qb = q + (size_t)batch * SEQ * EMB;
    const float* kb = k + (size_t)batch * SEQ * EMB;
    const float* vb = v + (size_t)batch * SEQ * EMB;

    // 25 dot products over 1024 dims, partitioned across 32 lanes
    float e[SEQ][SEQ];
#pragma unroll
    for (int i = 0; i < SEQ; ++i)
#pragma unroll
        for (int j = 0; j < SEQ; ++j) e[i][j] = 0.0f;

    for (int c = 0; c < EMB / 32; ++c) {
        int d = c * 32 + lane;
        float qr[SEQ], kr[SEQ];
#pragma unroll
        for (int i = 0; i < SEQ; ++i) { qr[i] = qb[i * EMB + d]; kr[i] = kb[i * EMB + d]; }
#pragma unroll
        for (int i = 0; i < SEQ; ++i)
#pragma unroll
            for (int j = 0; j < SEQ; ++j) e[i][j] = fmaf(qr[i], kr[j], e[i][j]);
    }
    // butterfly reduce across the wave32 (all lanes end up with the full sums)
#pragma unroll
    for (int i = 0; i < SEQ; ++i)
#pragma unroll
        for (int j = 0; j < SEQ; ++j) {
#pragma unroll
            for (int off = 16; off > 0; off >>= 1)
                e[i][j] += __shfl_xor(e[i][j], off, 32);
        }

    // time mixing: e1 = w_t @ e
    float wt[SEQ][SEQ];
#pragma unroll
    for (int a = 0; a < SEQ; ++a)
#pragma unroll
        for (int l = 0; l < SEQ; ++l) wt[a][l] = w_t[a * SEQ + l];
    float e1[SEQ][SEQ];
#pragma unroll
    for (int a = 0; a < SEQ; ++a)
#pragma unroll
        for (int vv = 0; vv < SEQ; ++vv) {
            float s = 0.0f;
#pragma unroll
            for (int l = 0; l < SEQ; ++l) s = fmaf(wt[a][l], e[l][vv], s);
            e1[a][vv] = s;
        }

    float g[SEQ], bt[SEQ];
#pragma unroll
    for (int i = 0; i < SEQ; ++i) { g[i] = gamma[i]; bt[i] = beta[i]; }

    // LayerNorm over last axis (5) + scaled softmax
    float att[SEQ][SEQ];
#pragma unroll
    for (int a = 0; a < SEQ; ++a) {
        float mu = 0.0f;
#pragma unroll
        for (int vv = 0; vv < SEQ; ++vv) mu += e1[a][vv];
        mu *= 0.2f;
        float var = 0.0f;
#pragma unroll
        for (int vv = 0; vv < SEQ; ++vv) { float d0 = e1[a][vv] - mu; var = fmaf(d0, d0, var); }
        var *= 0.2f;
        float r = rsqrtf(var + EPS);
        float s[SEQ];
        float mx = -3.4e38f;
#pragma unroll
        for (int vv = 0; vv < SEQ; ++vv) {
            s[vv] = ((e1[a][vv] - mu) * r * g[vv] + bt[vv]) * 0.03125f;  // / sqrt(1024)
            mx = fmaxf(mx, s[vv]);
        }
        float sum = 0.0f;
#pragma unroll
        for (int vv = 0; vv < SEQ; ++vv) { s[vv] = expf(s[vv] - mx); sum += s[vv]; }
        float inv = 1.0f / sum;
#pragma unroll
        for (int vv = 0; vv < SEQ; ++vv) att[a][vv] = s[vv] * inv;
    }

    // out = att @ v, streamed in 32-wide column chunks
    float* ob = out + (size_t)batch * SEQ * EMB;
    for (int c = 0; c < EMB / 32; ++c) {
        int d = c * 32 + lane;
        float vr[SEQ];
#pragma unroll
        for (int j = 0; j < SEQ; ++j) vr[j] = vb[j * EMB + d];
#pragma unroll
        for (int i = 0; i < SEQ; ++i) {
            float o = 0.0f;
#pragma unroll
            for (int j = 0; j < SEQ; ++j) o = fmaf(att[i][j], vr[j], o);
            ob[i * EMB + d] = o;
        }
    }
}

// ---------------------------------------------------------------------------
extern "C" void kernel_launch(void* const* d_in, const int* in_sizes, int n_in,
                              void* d_out, int out_size, void* d_ws, size_t ws_size,
                              hipStream_t stream) {
    const float* x     = (const float*)d_in[0];
    const float* w_t   = (const float*)d_in[1];
    const float* Wq    = (const float*)d_in[2];
    const float* bq    = (const float*)d_in[3];
    const float* Wk    = (const float*)d_in[4];
    const float* bk    = (const float*)d_in[5];
    const float* Wv    = (const float*)d_in[6];
    const float* bv    = (const float*)d_in[7];
    const float* gamma = (const float*)d_in[8];
    const float* beta  = (const float*)d_in[9];
    float* out = (float*)d_out;

    float* q = (float*)d_ws;
    float* k = q + (size_t)ROWS * EMB;
    float* v = k + (size_t)ROWS * EMB;

    dim3 grid1(ROWS / BM, 24);   // 320 x (3 matrices * 8 column-blocks)
    qkv_gemm_kernel<<<grid1, 256, 0, stream>>>(x, Wq, Wk, Wv, bq, bk, bv, q, k, v);

    attn_kernel<<<NBAT / 8, 256, 0, stream>>>(q, k, v, w_t, gamma, beta, out);
}